// Dwm_6244882448648
// MI455X (gfx1250) — compile-verified
//
#include <hip/hip_runtime.h>

// dwm3d / dense 3D conv (NDHWC, C=K=16, 3x3x3, SAME) on gfx1250 via
// V_WMMA_F32_16X16X4_F32 (full fp32 precision, matches reference numerics).
//
// Roofline: 32.6 GFLOP, >=302 MB HBM traffic (~13 us @ 23.3 TB/s); the whole
// input is L2-resident (151 MB < 192 MB). At fp32-WMMA rates the kernel is
// compute/LDS-feed bound, so the design maximizes WMMA per LDS byte:
//  * 3 M-tiles (48 output positions) per wave -> every B fragment feeds 3
//    WMMAs (10.7 B/lane/WMMA from LDS).
//  * B staged TRANSPOSED in LDS: lane's two B rows adjacent as float2 ->
//    one ds_load_b64 per B fragment (merges to 2addr_b64 across chunks),
//    hitting all 64 banks exactly once.
//  * A tiles 1280 B apart -> first two merge into ds_load_2addr_b64.
//  * Workgroup (4 waves / 128 thr) covers 2 full H-rows; 3(d) x 4(h) x 98(w)
//    zero-padded halo shared across rows (6x input amplification, all L2).
// Static LDS = 94,080 (x, stride-20 padded: 16B-aligned staging, conflict-
// free strided reads) + 27,648 (w) = 121,728 B -> 2 WG/WGP (<= 320 KB).
// 324 WMMAs per wave into three 8-VGPR accumulators.

typedef __attribute__((ext_vector_type(2))) float v2f;
typedef __attribute__((ext_vector_type(4))) float v4f;
typedef __attribute__((ext_vector_type(8))) float v8f;

#define Bn 8
#define Dn 32
#define Hn 96
#define Wn 96
#define Cn 16
#define Kn 16
#define NTAPS 27

#define HROWS 2               // output rows per workgroup
#define HHALO (HROWS + 2)     // 4 staged h-rows
#define HALO_W (Wn + 2)       // 98 padded positions
#define XSTRIDE 20            // floats per position in LDS
#define NTHREADS 128

__global__ __launch_bounds__(NTHREADS) void dwm3d_wmma(const float* __restrict__ x,
                                                       const float* __restrict__ wgt,
                                                       float* __restrict__ y)
{
    __shared__ float lx[3 * HHALO * HALO_W * XSTRIDE]; // 94,080 B
    __shared__ float lw[NTAPS * Cn * Kn];              // 27,648 B (transposed)

    const int tid = threadIdx.x;

    // blockIdx.x = (b*Dn + d)*(Hn/2) + hp ; h0 = 2*hp
    int idx = (int)blockIdx.x;
    const int hp = idx % (Hn / HROWS); idx /= (Hn / HROWS);
    const int d = idx % Dn; idx /= Dn;
    const int b = idx;
    const int h0 = hp * HROWS;

    // ---- stage weights TRANSPOSED:
    // lwT flat = tap*256 + q*64 + half*32 + n*2 + j  holds w[tap][4q+2*half+j][n]
    {
        for (int i = tid; i < NTAPS * 4 * 2 * Kn; i += NTHREADS) { // 3456 pairs
            const int n    = i & 15;
            const int half = (i >> 4) & 1;
            const int q    = (i >> 5) & 3;
            const int tap  = i >> 7;
            const int c    = 4 * q + 2 * half;
            v2f pair;
            pair.x = wgt[tap * (Cn * Kn) + c * Kn + n];
            pair.y = wgt[tap * (Cn * Kn) + (c + 1) * Kn + n];
            *(v2f*)(lw + i * 2) = pair;
        }
    }

    // ---- stage zero-padded input halo [3][HHALO][HALO_W][16] (stride 20) ----
    {
        const int nchunk = 3 * HHALO * HALO_W * 4; // float4 chunks = 4704
        for (int i = tid; i < nchunk; i += NTHREADS) {
            int cq = i & 3;
            int t = i >> 2;
            int p  = t % HALO_W; t /= HALO_W;
            int hr = t % HHALO;
            int dd = t / HHALO;
            int zd = d + dd - 1;
            int zh = h0 + hr - 1;
            int zw = p - 1;
            v4f val = {0.f, 0.f, 0.f, 0.f};
            if (zd >= 0 && zd < Dn && zh >= 0 && zh < Hn && zw >= 0 && zw < Wn) {
                const v4f* src =
                    (const v4f*)(x + ((((long)b * Dn + zd) * Hn + zh) * Wn + zw) * Cn);
                val = src[cq];
            }
            *(v4f*)(lx + ((dd * HHALO + hr) * HALO_W + p) * XSTRIDE + cq * 4) = val;
        }
    }
    __syncthreads();

    const int wid  = tid >> 5;        // wave 0..3
    const int lane = tid & 31;
    const int m    = lane & 15;       // A row / D column / B column
    const int half = lane >> 4;       // K/M half selector
    const int hloc = wid >> 1;        // local output row 0..1
    const int w0   = (wid & 1) * 48;  // output-position base (three 16-tiles)

    v8f acc0 = {};   // positions w0      .. w0+15
    v8f acc1 = {};   // positions w0+16   .. w0+31
    v8f acc2 = {};   // positions w0+32   .. w0+47

#pragma unroll
    for (int dd = 0; dd < 3; ++dd) {
#pragma unroll
      for (int hh = 0; hh < 3; ++hh) {
        const float* xrow = lx + (dd * HHALO + (hloc + hh)) * (HALO_W * XSTRIDE);
#pragma unroll
        for (int kw = 0; kw < 3; ++kw) {
          const int tap = (dd * 3 + hh) * 3 + kw;
          // A: positions (w0 + m + kw) + {0,16,32}; channels c0+2*half..+1
          const float* xpos = xrow + (w0 + m + kw) * XSTRIDE + 2 * half;
          // B (transposed): {w[c0+2*half][m], w[c0+2*half+1][m]} contiguous
          const float* wtap = lw + tap * 256 + half * 32 + m * 2;
#pragma unroll
          for (int c0 = 0; c0 < Cn; c0 += 4) {
            v2f a0 = *(const v2f*)(xpos + c0);
            v2f a1 = *(const v2f*)(xpos + 16 * XSTRIDE + c0);
            v2f a2 = *(const v2f*)(xpos + 32 * XSTRIDE + c0);
            v2f bb = *(const v2f*)(wtap + (c0 >> 2) * 64);    // ds_load_b64
            acc0 = __builtin_amdgcn_wmma_f32_16x16x4_f32(
                false, a0, false, bb, (short)0, acc0, false, false);
            acc1 = __builtin_amdgcn_wmma_f32_16x16x4_f32(
                false, a1, false, bb, (short)0, acc1, false, false);
            acc2 = __builtin_amdgcn_wmma_f32_16x16x4_f32(
                false, a2, false, bb, (short)0, acc2, false, false);
          }
        }
      }
    }

    // ---- store: lane holds N = m; rows r + 8*half of each 16-tile ----
    float* ybase =
        y + ((((long)b * Dn + d) * Hn + (h0 + hloc)) * Wn + w0) * Kn;
#pragma unroll
    for (int r = 0; r < 8; ++r) {
        const int mrow = r + 8 * half;
        ybase[(long)mrow * Kn + m]        = acc0[r];
        ybase[(long)(mrow + 16) * Kn + m] = acc1[r];
        ybase[(long)(mrow + 32) * Kn + m] = acc2[r];
    }
}

extern "C" void kernel_launch(void* const* d_in, const int* in_sizes, int n_in,
                              void* d_out, int out_size, void* d_ws, size_t ws_size,
                              hipStream_t stream) {
    const float* x = (const float*)d_in[0];
    const float* w = (const float*)d_in[1];
    float* y = (float*)d_out;
    dim3 grid(Bn * Dn * (Hn / HROWS)); // 12,288 workgroups
    dwm3d_wmma<<<grid, NTHREADS, 0, stream>>>(x, w, y);
}